// masked_softmax_sliding_window_6674379178452
// MI455X (gfx1250) — compile-verified
//
#include <hip/hip_runtime.h>
#include <stdint.h>

// Masked sliding-window softmax, MI455X (gfx1250, wave32).
//
// out[b,i,j] = softmax_j( mask(i,j) ? X[b,i,j] : -1e7 )
// mask(i,j): start(i) <= j < start(i)+200, start(i) = min(i,188)*30.
// In fp32 the masked entries are exactly 0 after softmax, so:
//   - read ONLY the 200-wide window per row  (~18.7 MB total reads)
//   - write softmax(window) + zeros elsewhere (~546 MB writes)
// Store-bandwidth bound: ~546MB / 23.3TB/s ~ 23us floor. Use 16B
// non-temporal stores for the zero fill, async global->LDS copy
// (ASYNCcnt path) for the window, ds_swizzle wave32 reductions.

#define MAXVAL   5843
#define TOPK     200
#define STEPSZ   30
#define NSTEPS   189          // ceil((5843-200)/30)
#define NTHREADS 256

typedef __attribute__((ext_vector_type(4))) float v4f;

// wave32 xor-lane shuffle via LDS swizzle unit (no LDS memory touched).
// group-of-32 pattern: and_mask=0x1f, or_mask=0, xor_mask=m
#define LANE_XOR_F(x, m)                                                      \
  __int_as_float(__builtin_amdgcn_ds_swizzle(__float_as_int(x),               \
                                             (0x1f | ((m) << 10))))

__device__ __forceinline__ void zero_fill_nt(float* __restrict__ p, int n,
                                             int tid) {
  if (n <= 0) return;
  // scalar head up to 16B alignment (p is always 4B aligned)
  int head = (int)(((16u - ((uint32_t)(uintptr_t)p & 15u)) & 15u) >> 2);
  if (head > n) head = n;
  for (int i = tid; i < head; i += NTHREADS)
    __builtin_nontemporal_store(0.0f, p + i);
  int rem = n - head;
  int n4  = rem >> 2;
  v4f* p4 = (v4f*)(p + head);
  v4f  z  = {0.0f, 0.0f, 0.0f, 0.0f};
  for (int i = tid; i < n4; i += NTHREADS)
    __builtin_nontemporal_store(z, p4 + i);
  int tail = head + (n4 << 2);
  for (int i = tail + tid; i < n; i += NTHREADS)
    __builtin_nontemporal_store(0.0f, p + i);
}

__global__ void __launch_bounds__(NTHREADS)
masked_softmax_sliding_window_kernel(const float* __restrict__ X,
                                     float* __restrict__ out) {
  const int row = blockIdx.x;   // query row 0..5842
  const int b   = blockIdx.y;   // batch 0..3
  const int tid = threadIdx.x;

  // dynamic LDS only => LDS byte offsets start at 0 (needed for async copy)
  extern __shared__ float smem[];          // [0..255]  window values
  float* red = smem + NTHREADS;            // [256..263] reduction scratch

  const int    sIdx   = (row < NSTEPS - 1) ? row : (NSTEPS - 1);
  const int    start  = sIdx * STEPSZ;     // window = [start, start+200)
  const size_t rowOff = ((size_t)b * MAXVAL + row) * (size_t)MAXVAL;
  const float* rowIn  = X + rowOff;
  float*       rowOut = out + rowOff;

  // ---- stage window into LDS with the CDNA5 async global->LDS path ----
  if (tid < TOPK) {
    uint32_t ldsOff = (uint32_t)(tid * 4);             // dyn-LDS byte offset
    uint32_t gOff   = (uint32_t)((start + tid) * 4);   // byte offset in row
    // GVS mode: dword at SADDR + VADDR goes to LDS byte offset VDST; tracked by ASYNCcnt
    asm volatile("global_load_async_to_lds_b32 %0, %1, %2"
                 :: "v"(ldsOff), "v"(gOff), "s"(rowIn)
                 : "memory");
  }
  asm volatile("s_wait_asynccnt 0" ::: "memory");
  __syncthreads();

  // ---- row max (wave32 swizzle tree + cross-wave LDS) ----
  float val = (tid < TOPK) ? smem[tid] : -3.0e38f;
  float m = val;
  m = fmaxf(m, LANE_XOR_F(m, 16));
  m = fmaxf(m, LANE_XOR_F(m, 8));
  m = fmaxf(m, LANE_XOR_F(m, 4));
  m = fmaxf(m, LANE_XOR_F(m, 2));
  m = fmaxf(m, LANE_XOR_F(m, 1));
  if ((tid & 31) == 0) red[tid >> 5] = m;
  __syncthreads();
  m = red[0];
#pragma unroll
  for (int i = 1; i < NTHREADS / 32; ++i) m = fmaxf(m, red[i]);

  // ---- exp + row sum ----
  float p = (tid < TOPK) ? __expf(val - m) : 0.0f;
  float s = p;
  s += LANE_XOR_F(s, 16);
  s += LANE_XOR_F(s, 8);
  s += LANE_XOR_F(s, 4);
  s += LANE_XOR_F(s, 2);
  s += LANE_XOR_F(s, 1);
  __syncthreads();                  // everyone done reading red[] (max pass)
  if ((tid & 31) == 0) red[tid >> 5] = s;
  __syncthreads();
  s = red[0];
#pragma unroll
  for (int i = 1; i < NTHREADS / 32; ++i) s += red[i];

  // ---- stores: zeros left, softmax window, zeros right (disjoint) ----
  zero_fill_nt(rowOut, start, tid);
  if (tid < TOPK) rowOut[start + tid] = p / s;
  zero_fill_nt(rowOut + start + TOPK, MAXVAL - start - TOPK, tid);
}

extern "C" void kernel_launch(void* const* d_in, const int* in_sizes, int n_in,
                              void* d_out, int out_size, void* d_ws, size_t ws_size,
                              hipStream_t stream) {
  (void)in_sizes; (void)n_in; (void)d_ws; (void)ws_size; (void)out_size;
  const float* X = (const float*)d_in[0];
  float* out = (float*)d_out;
  dim3 grid(MAXVAL, 4);
  dim3 block(NTHREADS);
  size_t smemBytes = (NTHREADS + NTHREADS / 32) * sizeof(float);
  masked_softmax_sliding_window_kernel<<<grid, block, smemBytes, stream>>>(X, out);
}